// HAPLoss_87471303951067
// MI455X (gfx1250) — compile-verified
//
#include <hip/hip_runtime.h>
#include <hip/hip_bf16.h>

#define BN 384
#define DIM 512
#define NLV 3

typedef __attribute__((ext_vector_type(2))) float v2f;
typedef __attribute__((ext_vector_type(8))) float v8f;

// ---------------------------------------------------------------------------
// Phase 1: scores = E @ E^T via V_WMMA_F32_16X16X4_F32 (exact fp32 numerics).
// One wave per 16x16 C tile; 24x24 = 576 tiles; 4 waves (128 thr) per block.
// A fragment (16x4 f32): lane<16 -> K={k,k+1}, lane>=16 -> K={k+2,k+3}, M=lane%16
// B fragment (4x16 f32): same K split, N=lane%16.  B(k,n) = E[j0+n][k].
// ---------------------------------------------------------------------------
#if __has_builtin(__builtin_amdgcn_wmma_f32_16x16x4_f32)
#define HAVE_WMMA_F32 1
#endif

__global__ __launch_bounds__(128) void gemm_scores_kernel(
    const float* __restrict__ E, float* __restrict__ S) {
  const int wave = threadIdx.x >> 5;
  const int lane = threadIdx.x & 31;
  const int tile = blockIdx.x * 4 + wave;        // 0..575
  const int ti = tile / 24, tj = tile % 24;
  const int i0 = ti * 16, j0 = tj * 16;
  const int half = lane >> 4;                    // 0: K={0,1}, 1: K={2,3}
  const int m = lane & 15;                       // row of A / col of B

  const float* arow = E + (size_t)(i0 + m) * DIM;
  const float* brow = E + (size_t)(j0 + m) * DIM;

#if HAVE_WMMA_F32
  v8f c = {};
#pragma unroll 4
  for (int k = 0; k < DIM; k += 4) {
    const int kk = k + 2 * half;                 // 8B-aligned (kk even)
    v2f a = *(const v2f*)(arow + kk);
    v2f b = *(const v2f*)(brow + kk);
    // (neg_a, A, neg_b, B, c_mod, C, reuse_a, reuse_b)
    c = __builtin_amdgcn_wmma_f32_16x16x4_f32(
        false, a, false, b, (short)0, c, false, false);
  }
#pragma unroll
  for (int r = 0; r < 8; ++r) {
    // C/D layout: VGPR r -> M = r + 8*half, N = m
    S[(size_t)(i0 + r + 8 * half) * BN + (j0 + m)] = c[r];
  }
#else
  // Fallback (should not be needed on gfx1250): scalar dot per C element.
  float c[8];
#pragma unroll
  for (int r = 0; r < 8; ++r) c[r] = 0.f;
  for (int k = 0; k < DIM; ++k) {
    float bv = E[(size_t)(j0 + m) * DIM + k];
#pragma unroll
    for (int r = 0; r < 8; ++r)
      c[r] += E[(size_t)(i0 + r + 8 * half) * DIM + k] * bv;
  }
#pragma unroll
  for (int r = 0; r < 8; ++r)
    S[(size_t)(i0 + r + 8 * half) * BN + (j0 + m)] = c[r];
#endif
}

// ---------------------------------------------------------------------------
// Phase 2: per-b row pass over the B^3 tensor. One block per b.
// s_row / t_row staged in LDS (3 KB); 8 waves own `i`, 32 lanes stride `j`,
// one exp per (b,i,j) element -> pure VALU/TRANS bound, zero HBM pressure.
// ---------------------------------------------------------------------------
__global__ __launch_bounds__(256) void hap_rows_kernel(
    const float* __restrict__ S, const int* __restrict__ labels,
    float* __restrict__ per_b) {
  const int b = blockIdx.x;
  __shared__ float s_row[BN];
  __shared__ float t_row[BN];
  __shared__ float wsum_h[8];
  __shared__ float wsum_r[8];

  const int tid = threadIdx.x;
  const int lane = tid & 31;
  const int wave = tid >> 5;

  const int lb0 = labels[b * NLV + 0];
  const int lb1 = labels[b * NLV + 1];
  const int lb2 = labels[b * NLV + 2];

  for (int j = tid; j < BN; j += 256) {
    s_row[j] = S[(size_t)b * BN + j];
    int cnt = (labels[j * NLV + 0] == lb0) + (labels[j * NLV + 1] == lb1) +
              (labels[j * NLV + 2] == lb2);
    t_row[j] = (float)cnt;
  }
  __syncthreads();

  float acc_hr = 0.f;   // sum_i hrank/rank  (lane 0 of each wave)
  float acc_rel = 0.f;  // sum_i relevances[b,i]

  for (int i = wave; i < BN; i += 8) {
    const float si = s_row[i];
    const float ti = t_row[i];
    const float reli = ti * (1.0f / 3.0f);
    float rsum = 0.f, hsum = 0.f;
    for (int j = lane; j < BN; j += 32) {
      if (j == i) continue;                      // mask = 1 - eye
      const float x = s_row[j] - si;             // sim_diff[b,i,j]
      const float tj = t_row[j];
      const float hv = (x < 0.f) ? 0.f : 1.f;    // heaviside
      float e = fminf(fmaxf(x * (-100.0f), -50.f), 50.f);  // clamp(-x/tau)
      const float sig = 1.0f / (1.0f + __expf(e));
      const float posb = (x > 0.05f) ? fmaf(100.f, x - 0.05f, 1.44f)
                                     : (0.5f + sig);
      const float infv = (x > 0.f) ? posb : sig;
      rsum += (ti > tj) ? infv : hv;             // step_rank
      hsum += hv * fminf(reli, tj * (1.0f / 3.0f));
    }
#pragma unroll
    for (int off = 16; off > 0; off >>= 1) {
      rsum += __shfl_down(rsum, off, 32);
      hsum += __shfl_down(hsum, off, 32);
    }
    if (lane == 0) {
      acc_hr += (reli + hsum) / (1.0f + rsum);
      acc_rel += reli;
    }
  }

  if (lane == 0) { wsum_h[wave] = acc_hr; wsum_r[wave] = acc_rel; }
  __syncthreads();
  if (tid == 0) {
    float h = 0.f, r = 0.f;
#pragma unroll
    for (int w = 0; w < 8; ++w) { h += wsum_h[w]; r += wsum_r[w]; }
    per_b[b] = 1.0f - h / r;                     // 1 - mhap[b]
  }
}

// ---------------------------------------------------------------------------
// Phase 3: deterministic single-block mean over 384 values.
// ---------------------------------------------------------------------------
__global__ __launch_bounds__(256) void final_reduce_kernel(
    const float* __restrict__ per_b, float* __restrict__ out) {
  __shared__ float red[256];
  const int tid = threadIdx.x;
  float a = 0.f;
  for (int i = tid; i < BN; i += 256) a += per_b[i];
  red[tid] = a;
  __syncthreads();
#pragma unroll
  for (int s = 128; s > 0; s >>= 1) {
    if (tid < s) red[tid] += red[tid + s];
    __syncthreads();
  }
  if (tid == 0) out[0] = red[0] * (1.0f / (float)BN);
}

extern "C" void kernel_launch(void* const* d_in, const int* in_sizes, int n_in,
                              void* d_out, int out_size, void* d_ws,
                              size_t ws_size, hipStream_t stream) {
  const float* emb = (const float*)d_in[0];        // [384, 512] f32
  const int* labels = (const int*)d_in[1];         // [384, 3] int
  float* S = (float*)d_ws;                         // [384, 384] scores
  float* per_b = (float*)((char*)d_ws + (size_t)BN * BN * sizeof(float));

  gemm_scores_kernel<<<576 / 4, 128, 0, stream>>>(emb, S);
  hap_rows_kernel<<<BN, 256, 0, stream>>>(S, labels, per_b);
  final_reduce_kernel<<<1, 256, 0, stream>>>(per_b, (float*)d_out);
}